// CropAndResize_1769526526006
// MI455X (gfx1250) — compile-verified
//
#include <hip/hip_runtime.h>

// crop_and_resize for image (8,256,200,200) f32, 1000 boxes, 14x14 crops.
// Pure gather/bandwidth kernel: ~200 MB out + ~0.8-3 GB gather in => no
// matmul structure (interp weights vary per output position), so the
// optimization target is data movement: coalesced NT stores, clustered
// 2x2 gathers, LDS-cached coordinates, and gfx1250 global_prefetch_b8
// one iteration ahead.

#define IMG_H   200
#define IMG_W   200
#define IMG_C   256
#define CROP_HH 14
#define CROP_WW 14
#define NPOS    (CROP_HH * CROP_WW)        // 196
#define PER_BOX (IMG_C * NPOS)             // 50176
#define PLANE   (IMG_H * IMG_W)            // 40000

typedef float v4f __attribute__((ext_vector_type(4)));

struct alignas(16) Coord {
    int   off;    // top_y * W + left_x (clamped so off+W+1 is in-bounds)
    float ly;     // y lerp weight
    float lx;     // x lerp weight
    float valid;  // 1.0 if inside image, else 0.0
};

__global__ __launch_bounds__(256)
void crop_and_resize_kernel(const float* __restrict__ image,
                            const float* __restrict__ boxes,
                            const int*   __restrict__ box_idx,
                            float*       __restrict__ out)
{
    __shared__ Coord sc[NPOS];

    const int b = blockIdx.x;      // box id (uniform -> scalar loads below)
    const int t = threadIdx.x;

    const float y1 = boxes[4 * b + 0];
    const float x1 = boxes[4 * b + 1];
    const float y2 = boxes[4 * b + 2];
    const float x2 = boxes[4 * b + 3];
    const float h_scale = (y2 - y1) * 199.0f / 13.0f;
    const float w_scale = (x2 - x1) * 199.0f / 13.0f;
    const float y_base  = y1 * 199.0f;
    const float x_base  = x1 * 199.0f;

    const int n = box_idx[b];
    const float* __restrict__ img = image + (size_t)n * (size_t)(IMG_C * PLANE);

    // ---- Stage 1: per-box coordinate table (reused by all 256 channels) ----
    if (t < NPOS) {
        const int py = t / CROP_WW;
        const int px = t - py * CROP_WW;
        float in_y = y_base + (float)py * h_scale;
        float in_x = x_base + (float)px * w_scale;

        const bool bad_y = (in_y > 199.0f) || (in_y < 0.0f);
        const bool bad_x = (in_x > 199.0f) || (in_x < 0.0f);
        if (bad_y) in_y = 0.0f;
        if (bad_x) in_x = 0.0f;

        int ty = (int)floorf(in_y);
        int lx = (int)floorf(in_x);
        // Clamp so the 2x2 patch stays in-bounds; lerp weight absorbs it
        // (only reachable when in_y/in_x land exactly on the last pixel).
        if (ty > IMG_H - 2) ty = IMG_H - 2;
        if (lx > IMG_W - 2) lx = IMG_W - 2;

        Coord c;
        c.off   = ty * IMG_W + lx;
        c.ly    = in_y - (float)ty;
        c.lx    = in_x - (float)lx;
        c.valid = (bad_y || bad_x) ? 0.0f : 1.0f;
        sc[t] = c;                       // ds_store_b128
    }
    __syncthreads();

    // ---- Stage 2: gather + bilinear over flattened (channel, pos) ----
    // PER_BOX/4 = 12544 float4 groups; 196 % 4 == 0 so a float4 never
    // straddles a channel boundary. 12544/256 = 49 exact iterations.
    float* __restrict__ outb = out + (size_t)b * PER_BOX;

    for (int k = 0; k < 49; ++k) {
        const int i4  = t + (k << 8);        // group index 0..12543
        const int c   = i4 / 49;             // channel 0..255
        const int pg  = i4 - c * 49;         // group within channel
        const int pos = pg << 2;             // sample index 0..192

        const float* __restrict__ plane = img + (size_t)c * PLANE;

        // gfx1250 path: warm L2/WGP$ for the NEXT iteration's 2x2 patch
        // lines (top row + bottom row, 800 B apart). Addresses are exactly
        // computable one iteration ahead; issue slots are free on a
        // bandwidth-bound kernel.
        if (k < 48) {
            const int i4n = t + ((k + 1) << 8);
            const int cn  = i4n / 49;
            const int pgn = i4n - cn * 49;
            const int offn = sc[pgn << 2].off;
            const float* pn = img + (size_t)cn * PLANE + offn;
            __builtin_prefetch(pn, 0, 3);             // global_prefetch_b8
            __builtin_prefetch(pn + IMG_W, 0, 3);     // global_prefetch_b8
        }

        v4f r;
        #pragma unroll
        for (int j = 0; j < 4; ++j) {
            const Coord cd = sc[pos + j];            // ds_load_b128
            const float* p = plane + cd.off;
            const float tl = p[0];                   // imm offsets 0,4,800,804
            const float tr = p[1];
            const float bl = p[IMG_W];
            const float br = p[IMG_W + 1];
            const float top = fmaf(cd.lx, tr - tl, tl);
            const float bot = fmaf(cd.lx, br - bl, bl);
            const float v   = fmaf(cd.ly, bot - top, top);
            r[j] = v * cd.valid;
        }
        // Streaming output: NT hint keeps L2 for the gathered image data.
        __builtin_nontemporal_store(r, (v4f*)(outb + (size_t)i4 * 4));
    }
}

extern "C" void kernel_launch(void* const* d_in, const int* in_sizes, int n_in,
                              void* d_out, int out_size, void* d_ws, size_t ws_size,
                              hipStream_t stream) {
    const float* image   = (const float*)d_in[0];
    const float* boxes   = (const float*)d_in[1];
    const int*   box_idx = (const int*)d_in[2];
    float*       out     = (float*)d_out;

    const int nboxes = in_sizes[1] / 4;   // 1000
    crop_and_resize_kernel<<<dim3(nboxes), dim3(256), 0, stream>>>(
        image, boxes, box_idx, out);
}